// GraphRegressor_67095979098760
// MI455X (gfx1250) — compile-verified
//
#include <hip/hip_runtime.h>

#define HDIM 128

typedef float v2f __attribute__((ext_vector_type(2)));
typedef float v8f __attribute__((ext_vector_type(8)));

// ---------------- utility kernels ----------------

__global__ void set_val_f32(float* __restrict__ p, float v, size_t n) {
  size_t i = (size_t)blockIdx.x * blockDim.x + threadIdx.x;
  if (i < n) p[i] = v;
}

// deg[dst[e]] += 1  (deg pre-initialized to 1.0 == self loop)
__global__ void count_deg(const int* __restrict__ dst, float* __restrict__ deg, int E) {
  int e = blockIdx.x * blockDim.x + threadIdx.x;
  if (e < E) atomicAdd(&deg[dst[e]], 1.0f);
}

// ---------------- WMMA GEMM: C[M x 128] = A[M x 128] @ W[128 x 128] + bias ----------------
// One wave computes one 16x16 output tile; 8 waves/block cover 16 rows x 128 cols.
// f32 WMMA (16x16x4) keeps exact f32 math; GEMM is ~10 GFLOP total, not the bottleneck.
__global__ void gemm128_wmma(const float* __restrict__ A, const float* __restrict__ W,
                             const float* __restrict__ bias, float* __restrict__ C,
                             int M, int relu) {
  const int lane = threadIdx.x & 31;
  const int wave = threadIdx.x >> 5;    // 0..7 -> 16-col tile
  const int row0 = blockIdx.x * 16;
  const int col0 = wave * 16;
  const int half = lane >> 4;           // 0: K pair {k,k+1}, 1: {k+2,k+3}
  const int l16  = lane & 15;

  int arow = row0 + l16;                // clamp so EXEC stays all-ones through WMMA
  if (arow > M - 1) arow = M - 1;
  const float* __restrict__ Ap = A + (size_t)arow * HDIM;

  v8f acc = {};
#pragma unroll
  for (int k = 0; k < HDIM; k += 4) {
    const int ka = k + 2 * half;
    v2f a, b;
    a[0] = Ap[ka];
    a[1] = Ap[ka + 1];
    b[0] = W[(size_t)ka * HDIM + col0 + l16];
    b[1] = W[(size_t)(ka + 1) * HDIM + col0 + l16];
    acc = __builtin_amdgcn_wmma_f32_16x16x4_f32(false, a, false, b, (short)0, acc,
                                                false, false);
  }

  const float bv = bias[col0 + l16];
#pragma unroll
  for (int r = 0; r < 8; ++r) {
    const int row = row0 + r + 8 * half;   // C/D layout: VGPR r -> M=r (lanes 0-15) / M=r+8
    if (row < M) {
      float v = acc[r] + bv;
      if (relu) v = fmaxf(v, 0.0f);
      C[(size_t)row * HDIM + col0 + l16] = v;
    }
  }
}

// ---------------- edge scatter: acc[dst] += h[src] * rsqrt(deg[s]*deg[d]) ----------------
// One wave per edge; each lane handles 4 contiguous floats (coalesced 512B row).
// h/acc are 51.2MB each -> L2 resident (192MB), atomics resolve in L2.
__global__ void scatter_edges(const int* __restrict__ src, const int* __restrict__ dst,
                              const float* __restrict__ h, const float* __restrict__ deg,
                              float* __restrict__ acc, int E) {
  const int wid  = (int)(((size_t)blockIdx.x * blockDim.x + threadIdx.x) >> 5);
  const int lane = threadIdx.x & 31;
  if (wid >= E) return;
  const int s = src[wid];
  const int d = dst[wid];
  const float nrm = rsqrtf(deg[s] * deg[d]);   // deg >= 1 always (self loops)
  const float4 hv = *(const float4*)(h + (size_t)s * HDIM + lane * 4);
  float* ad = acc + (size_t)d * HDIM + lane * 4;
  atomicAdd(ad + 0, hv.x * nrm);
  atomicAdd(ad + 1, hv.y * nrm);
  atomicAdd(ad + 2, hv.z * nrm);
  atomicAdd(ad + 3, hv.w * nrm);
}

// x_out = relu(acc + h / deg)   (self-loop norm = rsqrt(deg^2) = 1/deg)
__global__ void selfloop_relu(const float* __restrict__ h, const float* __restrict__ acc,
                              const float* __restrict__ deg, float* __restrict__ xout,
                              int total) {
  int i = blockIdx.x * blockDim.x + threadIdx.x;
  if (i >= total) return;
  const float v = acc[i] + h[i] / deg[i >> 7];
  xout[i] = fmaxf(v, 0.0f);
}

// pool[batch[i]] += x[i]; cnt[batch[i]] += 1   (one wave per node)
__global__ void pool_scatter(const float* __restrict__ x, const int* __restrict__ batch,
                             float* __restrict__ pool, float* __restrict__ cnt, int N) {
  const int wid  = (int)(((size_t)blockIdx.x * blockDim.x + threadIdx.x) >> 5);
  const int lane = threadIdx.x & 31;
  if (wid >= N) return;
  const int g = batch[wid];
  const float4 v = *(const float4*)(x + (size_t)wid * HDIM + lane * 4);
  float* p = pool + (size_t)g * HDIM + lane * 4;
  atomicAdd(p + 0, v.x);
  atomicAdd(p + 1, v.y);
  atomicAdd(p + 2, v.z);
  atomicAdd(p + 3, v.w);
  if (lane == 0) atomicAdd(cnt + g, 1.0f);
}

__global__ void pool_div(float* __restrict__ pool, const float* __restrict__ cnt, int total) {
  int i = blockIdx.x * blockDim.x + threadIdx.x;
  if (i >= total) return;
  float c = cnt[i >> 7];
  pool[i] = pool[i] / fmaxf(c, 1.0f);
}

// out[g] = dot(hid[g], Wp2) + bp2
__global__ void final_head(const float* __restrict__ hid, const float* __restrict__ Wp2,
                           const float* __restrict__ bp2, float* __restrict__ out, int G) {
  int g = blockIdx.x * blockDim.x + threadIdx.x;
  if (g >= G) return;
  const float* hr = hid + (size_t)g * HDIM;
  float s = bp2[0];
#pragma unroll 8
  for (int c = 0; c < HDIM; ++c) s += hr[c] * Wp2[c];
  out[g] = s;
}

// ---------------- launch ----------------

static inline int ceil_div_i(long long a, long long b) { return (int)((a + b - 1) / b); }

extern "C" void kernel_launch(void* const* d_in, const int* in_sizes, int n_in,
                              void* d_out, int out_size, void* d_ws, size_t ws_size,
                              hipStream_t stream) {
  const float* x     = (const float*)d_in[0];
  const int*   ei    = (const int*)d_in[1];
  const int*   batch = (const int*)d_in[2];
  const float* Wl[3] = {(const float*)d_in[3], (const float*)d_in[5], (const float*)d_in[7]};
  const float* bl[3] = {(const float*)d_in[4], (const float*)d_in[6], (const float*)d_in[8]};
  const float* Wp1   = (const float*)d_in[9];
  const float* bp1   = (const float*)d_in[10];
  const float* Wp2   = (const float*)d_in[11];
  const float* bp2   = (const float*)d_in[12];
  float* out = (float*)d_out;

  const int N  = in_sizes[0] / HDIM;     // 100000
  const int E  = in_sizes[1] / 2;        // 1600000
  const int G  = out_size;               // 1000
  const int Gp = (G + 15) & ~15;         // pad graphs to WMMA tile multiple

  const int* src = ei;
  const int* dst = ei + E;

  // workspace layout (floats), each region 256B-aligned
  auto align_f = [](size_t v) { return (v + 63) & ~(size_t)63; };
  float* ws = (float*)d_ws;
  size_t off = 0;
  float* deg  = ws + off; off += align_f((size_t)N);
  float* hbuf = ws + off; off += align_f((size_t)N * HDIM);
  float* acc  = ws + off; off += align_f((size_t)N * HDIM);
  float* xbuf = ws + off; off += align_f((size_t)N * HDIM);
  float* pool = ws + off; off += align_f((size_t)Gp * HDIM);
  float* cnt  = ws + off; off += align_f((size_t)Gp);
  float* hid  = ws + off; off += align_f((size_t)Gp * HDIM);
  (void)ws_size; (void)n_in;

  const int B = 256;
  const size_t NH = (size_t)N * HDIM;

  // 1) degrees (init to 1.0 for the self loop, then count incoming edges)
  set_val_f32<<<ceil_div_i(N, B), B, 0, stream>>>(deg, 1.0f, (size_t)N);
  count_deg<<<ceil_div_i(E, B), B, 0, stream>>>(dst, deg, E);

  // 2) three GCN layers
  const float* xin = x;
  for (int l = 0; l < 3; ++l) {
    gemm128_wmma<<<ceil_div_i(N, 16), B, 0, stream>>>(xin, Wl[l], bl[l], hbuf, N, 0);
    set_val_f32<<<ceil_div_i((long long)NH, B), B, 0, stream>>>(acc, 0.0f, NH);
    scatter_edges<<<ceil_div_i((long long)E * 32, B), B, 0, stream>>>(src, dst, hbuf, deg,
                                                                      acc, E);
    selfloop_relu<<<ceil_div_i((long long)NH, B), B, 0, stream>>>(hbuf, acc, deg, xbuf,
                                                                  (int)NH);
    xin = xbuf;
  }

  // 3) segment-mean pool (padded to Gp rows, zero-initialized)
  set_val_f32<<<ceil_div_i((long long)Gp * HDIM + Gp, B), B, 0, stream>>>(
      pool, 0.0f, (size_t)Gp * HDIM + (size_t)Gp);  // pool and cnt are contiguous
  pool_scatter<<<ceil_div_i((long long)N * 32, B), B, 0, stream>>>(xbuf, batch, pool, cnt, N);
  pool_div<<<ceil_div_i((long long)Gp * HDIM, B), B, 0, stream>>>(pool, cnt, Gp * HDIM);

  // 4) MLP head: hid = relu(pool @ Wp1 + bp1); out = hid @ Wp2 + bp2
  gemm128_wmma<<<ceil_div_i(Gp, 16), B, 0, stream>>>(pool, Wp1, bp1, hid, Gp, 1);
  final_head<<<ceil_div_i(G, B), B, 0, stream>>>(hid, Wp2, bp2, out, G);
}